// PyramidROIAlign_20212116095249
// MI455X (gfx1250) — compile-verified
//
#include <hip/hip_runtime.h>

// PyramidROIAlign for MI455X (gfx1250, wave32).
// One wave per output pixel (b, n, py, px); 256 channels per pixel.
// Bilinear blend computed as D = A(16x4 weights) x B(4x16 corner values)
// via V_WMMA_F32_16X16X4_F32 (full f32, K=4 == number of bilinear corners).
// Output stores are non-temporal so the streamed 100MB result does not evict
// the ~170MB feature-map working set from the 192MB global L2.

typedef __attribute__((ext_vector_type(2))) float v2f;
typedef __attribute__((ext_vector_type(8))) float v8f;

#define POOL 7
#define CCH  256
#define NBOX 1000
#define NB   2

__global__ __launch_bounds__(256) void roi_align_wmma_kernel(
    const float* __restrict__ boxes,       // [NB, NBOX, 4] (y1,x1,y2,x2)
    const float* __restrict__ image_meta,  // [NB, 12], [0,4:6] = (H, W)
    const float* __restrict__ p2,          // [NB,256,256,256]
    const float* __restrict__ p3,          // [NB,128,128,256]
    const float* __restrict__ p4,          // [NB, 64, 64,256]
    const float* __restrict__ p5,          // [NB, 32, 32,256]
    float* __restrict__ out)               // [NB, NBOX, 7, 7, 256]
{
    const int lane = threadIdx.x & 31;
    const int wid  = blockIdx.x * (blockDim.x >> 5) + (threadIdx.x >> 5);
    const int total = NB * NBOX * POOL * POOL;
    if (wid >= total) return;            // wave-uniform (grid is exact anyway)

    const int pix = wid % (POOL * POOL);
    const int bn  = wid / (POOL * POOL);
    const int n   = bn % NBOX;
    const int b   = bn / NBOX;
    const int py  = pix / POOL;
    const int px  = pix % POOL;

    // ---- box + pyramid level (wave-uniform scalar work) ----
    const float* box = boxes + ((size_t)b * NBOX + n) * 4;
    const float y1 = box[0], x1 = box[1], y2 = box[2], x2 = box[3];
    const float h = y2 - y1, w = x2 - x1;

    const float imgH = image_meta[4], imgW = image_meta[5];
    // scale = sqrt(h*w) / (224/sqrt(H*W)); level = clip(4 + round(log2(scale)), 2, 5)
    const float scale = sqrtf(h * w) * (sqrtf(imgH * imgW) / 224.0f);
    int level = 4 + (int)rintf(log2f(scale));   // rintf: round-half-even == jnp.round
    level = min(5, max(2, level));

    const float* fm; int Hf;
    switch (level) {
        case 2:  fm = p2; Hf = 256; break;
        case 3:  fm = p3; Hf = 128; break;
        case 4:  fm = p4; Hf = 64;  break;
        default: fm = p5; Hf = 32;  break;
    }
    const int Wf = Hf;

    // ---- sample position, clamp, validity, bilinear weights ----
    const float fy = (float)py / 6.0f;          // exact 1.0 at py==6
    const float fx = (float)px / 6.0f;
    const float in_y = (y1 + fy * h) * (float)(Hf - 1);
    const float in_x = (x1 + fx * w) * (float)(Wf - 1);
    const bool vy = (in_y >= 0.0f) && (in_y <= (float)(Hf - 1));
    const bool vx = (in_x >= 0.0f) && (in_x <= (float)(Wf - 1));
    const float mval = (vy && vx) ? 1.0f : 0.0f;

    const float y0f = floorf(in_y), x0f = floorf(in_x);
    const float dy = in_y - y0f,    dx = in_x - x0f;
    const int y0  = min(Hf - 1, max(0, (int)y0f));
    const int y1i = min(Hf - 1, max(0, (int)y0f + 1));
    const int x0  = min(Wf - 1, max(0, (int)x0f));
    const int x1i = min(Wf - 1, max(0, (int)x0f + 1));

    const float w_tl = (1.0f - dy) * (1.0f - dx) * mval;
    const float w_tr = (1.0f - dy) * dx          * mval;
    const float w_bl = dy          * (1.0f - dx) * mval;
    const float w_br = dy          * dx          * mval;

    // Corner channel-rows (each 256 contiguous f32 = 1 KB)
    const float* row_tl = fm + (((size_t)b * Hf + y0 ) * Wf + x0 ) * CCH;
    const float* row_tr = fm + (((size_t)b * Hf + y0 ) * Wf + x1i) * CCH;
    const float* row_bl = fm + (((size_t)b * Hf + y1i) * Wf + x0 ) * CCH;
    const float* row_br = fm + (((size_t)b * Hf + y1i) * Wf + x1i) * CCH;

    // ---- WMMA operand staging ----
    // A (16x4 f32): lane m in 0..15 holds K=0 (vgpr0), K=1 (vgpr1);
    //               lanes 16..31 hold K=2, K=3. Broadcast weights to all M rows.
    const bool lo = (lane < 16);
    v2f A;
    A[0] = lo ? w_tl : w_bl;   // K=0 -> tl, K=2 -> bl
    A[1] = lo ? w_tr : w_br;   // K=1 -> tr, K=3 -> br

    // B (4x16 f32): mirrored K split — lanes 0..15 supply K=0,1 rows,
    // lanes 16..31 supply K=2,3 rows, each lane-striped over 16 channels.
    const float* pB0 = lo ? row_tl : row_bl;
    const float* pB1 = lo ? row_tr : row_br;

    float* orow = out + (size_t)wid * CCH;
    const int cl = lane & 15;
    const v8f cacc = {0, 0, 0, 0, 0, 0, 0, 0};

    #pragma unroll
    for (int c0 = 0; c0 < CCH; c0 += 32) {
        v2f Ba, Bb;
        Ba[0] = pB0[c0 + cl];           // channels c0..c0+15
        Ba[1] = pB1[c0 + cl];
        Bb[0] = pB0[c0 + 16 + cl];      // channels c0+16..c0+31
        Bb[1] = pB1[c0 + 16 + cl];

        // D[m,n] = w_tl*tl[n] + w_tr*tr[n] + w_bl*bl[n] + w_br*br[n] (all m identical)
        v8f dA = __builtin_amdgcn_wmma_f32_16x16x4_f32(
            false, A, false, Ba, (short)0, cacc, false, false);
        v8f dB = __builtin_amdgcn_wmma_f32_16x16x4_f32(
            false, A, false, Bb, (short)0, cacc, false, false);

        // d[0]: lanes 0..15 = D[0, lane]; lanes 16..31 = D[8, lane-16] (same row value).
        // Merge the two 16-channel slabs into one coalesced 128B non-temporal store.
        const float val = lo ? dA[0] : dB[0];
        __builtin_nontemporal_store(val, &orow[c0 + lane]);
    }
}

extern "C" void kernel_launch(void* const* d_in, const int* in_sizes, int n_in,
                              void* d_out, int out_size, void* d_ws, size_t ws_size,
                              hipStream_t stream) {
    const float* boxes = (const float*)d_in[0];
    const float* meta  = (const float*)d_in[1];
    const float* p2    = (const float*)d_in[2];
    const float* p3    = (const float*)d_in[3];
    const float* p4    = (const float*)d_in[4];
    const float* p5    = (const float*)d_in[5];
    float* out = (float*)d_out;

    const int total = NB * NBOX * POOL * POOL;   // 98000 waves, one per pixel
    const int wavesPerBlock = 8;                 // 256 threads = 8 wave32
    const int blocks = (total + wavesPerBlock - 1) / wavesPerBlock;  // 12250
    roi_align_wmma_kernel<<<blocks, 256, 0, stream>>>(boxes, meta, p2, p3, p4, p5, out);
}